// Hybrid_20521353740367
// MI455X (gfx1250) — compile-verified
//
#include <hip/hip_runtime.h>
#include <math.h>

typedef __attribute__((ext_vector_type(2))) float v2f;
typedef __attribute__((ext_vector_type(8))) float v8f;

#define TPB 256
#define MT 2   // 16-row tiles per wave in the WMMA GEMM (B-fragment reuse)

static inline unsigned gblocks(long n) { return (unsigned)((n + TPB - 1) / TPB); }

// ---------------- elementwise utility kernels ----------------
__global__ void fill_kernel(float* __restrict__ p, float v, long n) {
  long i = (long)blockIdx.x * TPB + threadIdx.x;
  if (i < n) p[i] = v;
}
__global__ void copy_kernel(float* __restrict__ d, const float* __restrict__ s, long n) {
  long i = (long)blockIdx.x * TPB + threadIdx.x;
  if (i < n) d[i] = s[i];
}
__global__ void negcopy_kernel(float* __restrict__ d, const float* __restrict__ s, long n) {
  long i = (long)blockIdx.x * TPB + threadIdx.x;
  if (i < n) d[i] = -s[i];
}

// ---------------- encoder kernels ----------------
// direct 3x3 conv, SAME padding, NCHW, OIHW weights, optional relu
__global__ void conv3x3_kernel(const float* __restrict__ in, const float* __restrict__ w,
                               const float* __restrict__ bias, float* __restrict__ out,
                               int Cin, int Cout, int S, int relu) {
  long idx = (long)blockIdx.x * TPB + threadIdx.x;
  long total = 2L * Cout * S * S;
  if (idx >= total) return;
  int x = (int)(idx % S); long t = idx / S;
  int y = (int)(t % S); t /= S;
  int co = (int)(t % Cout); int b = (int)(t / Cout);
  float acc = bias[co];
  const float* wbase = w + (size_t)co * Cin * 9;
  const float* ibase = in + ((size_t)b * Cin) * S * S;
  for (int ci = 0; ci < Cin; ++ci) {
    const float* ip = ibase + (size_t)ci * S * S;
    const float* wp = wbase + ci * 9;
    for (int dy = 0; dy < 3; ++dy) {
      int yy = y + dy - 1;
      if (yy < 0 || yy >= S) continue;
      const float* row = ip + (size_t)yy * S;
      for (int dx = 0; dx < 3; ++dx) {
        int xx = x + dx - 1;
        if (xx < 0 || xx >= S) continue;
        acc += row[xx] * wp[dy * 3 + dx];
      }
    }
  }
  if (relu) acc = fmaxf(acc, 0.f);
  out[idx] = acc;
}

// out = relu(h2 + bs + conv1x1(x, ws))
__global__ void skip1x1_kernel(const float* __restrict__ xin, const float* __restrict__ h2,
                               const float* __restrict__ wsk, const float* __restrict__ bsk,
                               float* __restrict__ out, int Cin, int Cout, int S) {
  long idx = (long)blockIdx.x * TPB + threadIdx.x;
  long total = 2L * Cout * S * S;
  if (idx >= total) return;
  int x = (int)(idx % S); long t = idx / S;
  int y = (int)(t % S); t /= S;
  int co = (int)(t % Cout); int b = (int)(t / Cout);
  float acc = h2[idx] + bsk[co];
  const float* wp = wsk + (size_t)co * Cin;
  const float* ip = xin + ((size_t)b * Cin) * S * S + (size_t)y * S + x;
  for (int ci = 0; ci < Cin; ++ci) acc += ip[(size_t)ci * S * S] * wp[ci];
  out[idx] = fmaxf(acc, 0.f);
}

__global__ void pool2_kernel(const float* __restrict__ in, float* __restrict__ out,
                             int C, int S) {
  int O = S >> 1;
  long idx = (long)blockIdx.x * TPB + threadIdx.x;
  long total = 2L * C * O * O;
  if (idx >= total) return;
  int x = (int)(idx % O); long t = idx / O;
  int y = (int)(t % O); t /= O;
  int c = (int)(t % C); int b = (int)(t / C);
  const float* ip = in + (((size_t)b * C + c) * S + 2 * y) * S + 2 * x;
  out[idx] = fmaxf(fmaxf(ip[0], ip[1]), fmaxf(ip[S], ip[S + 1]));
}

// mu[b,o] = bias[o] + dot(flat[b,:], W[o,:]), In = 131072
__global__ void fc_kernel(const float* __restrict__ flat, const float* __restrict__ Wf,
                          const float* __restrict__ bf, float* __restrict__ mu, int In) {
  int o = blockIdx.x, b = blockIdx.y;
  const float* f = flat + (size_t)b * In;
  const float* w = Wf + (size_t)o * In;
  float s = 0.f;
  for (int i = threadIdx.x; i < In; i += TPB) s += f[i] * w[i];
  __shared__ float red[TPB];
  red[threadIdx.x] = s;
  __syncthreads();
  for (int st = TPB / 2; st > 0; st >>= 1) {
    if (threadIdx.x < st) red[threadIdx.x] += red[threadIdx.x + st];
    __syncthreads();
  }
  if (threadIdx.x == 0) mu[b * 64 + o] = red[0] + bf[o];
}

// g0[b,j] = relu(dec_b[j] + sum_l mu[b,l] * dec_w[j,l]), NJ = 160000
__global__ void decoder_kernel(const float* __restrict__ Wd, const float* __restrict__ bd,
                               const float* __restrict__ mu, float* __restrict__ g0, int NJ) {
  int j = blockIdx.x * TPB + threadIdx.x;
  if (j >= NJ) return;
  const float* wr = Wd + (size_t)j * 64;
  for (int b = 0; b < 2; ++b) {
    float s = bd[j];
    for (int l = 0; l < 64; ++l) s += mu[b * 64 + l] * wr[l];
    g0[(size_t)b * NJ + j] = fmaxf(s, 0.f);
  }
}

// ---------------- graph kernels ----------------
__global__ void deg_kernel(const int* __restrict__ dst, int E, float* __restrict__ deg) {
  int e = blockIdx.x * TPB + threadIdx.x;
  if (e < E) atomicAdd(&deg[dst[e]], 1.0f);
}
__global__ void dinv_kernel(const float* __restrict__ deg, float* __restrict__ dinv, int N) {
  int n = blockIdx.x * TPB + threadIdx.x;
  if (n < N) { float d = deg[n]; dinv[n] = (d > 0.f) ? (1.0f / sqrtf(d)) : 0.f; }
}

// out[:,dst,f] += scale * (-(dinv[src]*dinv[dst])) * in[:,src,f]   (B=2)
__global__ void prop_kernel(const float* __restrict__ in, float* __restrict__ out,
                            const int* __restrict__ src, const int* __restrict__ dst,
                            int E, const float* __restrict__ dinv, int F, int N, float scale) {
  long t = (long)blockIdx.x * TPB + threadIdx.x;
  if (t >= (long)E * F) return;
  int e = (int)(t / F), f = (int)(t % F);
  int s = src[e], d = dst[e];
  float nrm = -(dinv[s] * dinv[d]) * scale;
  atomicAdd(&out[((size_t)d) * F + f], in[((size_t)s) * F + f] * nrm);
  atomicAdd(&out[((size_t)(N + d)) * F + f], in[((size_t)(N + s)) * F + f] * nrm);
}

// C[M,G] += A[M,F] * W[F,G] via V_WMMA_F32_16X16X4_F32.
// One wave per 32x16 C tile (MT=2 row tiles share the B fragment).
// Interior tiles: fully unguarded loads/stores. Edge tiles / K-tail:
// clamp addresses + v_cndmask the values (no exec diamonds).
__global__ void wmma_gemm_acc(const float* __restrict__ A, const float* __restrict__ W,
                              float* __restrict__ C, int M, int F, int G) {
  int lane = threadIdx.x;          // 0..31, wave32
  int half = lane >> 4;            // 0: lanes 0-15, 1: lanes 16-31
  int l = lane & 15;
  long mbase = (long)blockIdx.x * (16 * MT);
  int nbase = blockIdx.y * 16;
  int col = nbase + l;
  int F4 = F & ~3;
#if defined(__HIP_DEVICE_COMPILE__) && __has_builtin(__builtin_amdgcn_wmma_f32_16x16x4_f32)
  v8f c[MT];
  bool interior = (mbase + 16 * MT <= M) && (nbase + 16 <= G);
  if (interior) {
    for (int t = 0; t < MT; ++t)
      for (int j = 0; j < 8; ++j)
        c[t][j] = C[(size_t)(mbase + t * 16 + j + 8 * half) * G + col];
    for (int k = 0; k < F4; k += 4) {
      int kk = k + 2 * half;                 // A: lane=M, K = 2*half + {0,1}
      v2f b;
      b.x = W[(size_t)kk * G + col];         // B: lane=N, same K pair
      b.y = W[(size_t)(kk + 1) * G + col];
      for (int t = 0; t < MT; ++t) {
        v2f a = *(const v2f*)(A + (size_t)(mbase + t * 16 + l) * F + kk);
        c[t] = __builtin_amdgcn_wmma_f32_16x16x4_f32(false, a, false, b, (short)0,
                                                     c[t], false, false);
      }
    }
    if (F4 < F) {                            // K tail (e.g. F=146): masked, branch-free
      int kk = F4 + 2 * half;
      int k0 = min(kk, F - 1), k1 = min(kk + 1, F - 1);
      v2f b;
      float bx = W[(size_t)k0 * G + col], by = W[(size_t)k1 * G + col];
      b.x = (kk < F) ? bx : 0.f;
      b.y = (kk + 1 < F) ? by : 0.f;
      for (int t = 0; t < MT; ++t) {
        const float* ar = A + (size_t)(mbase + t * 16 + l) * F;
        float ax = ar[k0], ay = ar[k1];
        v2f a;
        a.x = (kk < F) ? ax : 0.f;
        a.y = (kk + 1 < F) ? ay : 0.f;
        c[t] = __builtin_amdgcn_wmma_f32_16x16x4_f32(false, a, false, b, (short)0,
                                                     c[t], false, false);
      }
    }
    for (int t = 0; t < MT; ++t)
      for (int j = 0; j < 8; ++j)
        C[(size_t)(mbase + t * 16 + j + 8 * half) * G + col] = c[t][j];
  } else {
    // edge tiles (G=2 heads, row remainder): clamped loads, masked values
    bool cok = col < G;
    int colc = cok ? col : (G - 1);
    for (int t = 0; t < MT; ++t)
      for (int j = 0; j < 8; ++j) {
        long row = mbase + t * 16 + j + 8 * half;
        long rowc = (row < M) ? row : (M - 1);
        float v = C[(size_t)rowc * G + colc];
        c[t][j] = (row < M && cok) ? v : 0.f;
      }
    for (int k = 0; k < F; k += 4) {
      int kk = k + 2 * half;
      int k0 = min(kk, F - 1), k1 = min(kk + 1, F - 1);
      float bx = W[(size_t)k0 * G + colc], by = W[(size_t)k1 * G + colc];
      v2f b;
      b.x = (cok && kk < F) ? bx : 0.f;
      b.y = (cok && kk + 1 < F) ? by : 0.f;
      for (int t = 0; t < MT; ++t) {
        long row = mbase + t * 16 + l;
        long rowc = (row < M) ? row : (M - 1);
        const float* ar = A + (size_t)rowc * F;
        float ax = ar[k0], ay = ar[k1];
        v2f a;
        a.x = (row < M && kk < F) ? ax : 0.f;
        a.y = (row < M && kk + 1 < F) ? ay : 0.f;
        c[t] = __builtin_amdgcn_wmma_f32_16x16x4_f32(false, a, false, b, (short)0,
                                                     c[t], false, false);
      }
    }
    for (int t = 0; t < MT; ++t)
      for (int j = 0; j < 8; ++j) {
        long row = mbase + t * 16 + j + 8 * half;
        if (row < M && cok) C[(size_t)row * G + col] = c[t][j];
      }
  }
#else
  for (int t = 0; t < MT; ++t)
    for (int j = 0; j < 8; ++j) {
      long row = mbase + t * 16 + j + 8 * half;
      if (row < M && col < G) {
        float acc = C[(size_t)row * G + col];
        for (int k = 0; k < F; ++k) acc += A[(size_t)row * F + k] * W[(size_t)k * G + col];
        C[(size_t)row * G + col] = acc;
      }
    }
#endif
}

// per-row instance norm over last dim F, optional relu; rows = 2*N
__global__ void inorm_kernel(float* __restrict__ g, long rows, int F, int relu) {
  long r = (long)blockIdx.x * TPB + threadIdx.x;
  if (r >= rows) return;
  float* p = g + r * (size_t)F;
  float s = 0.f, s2 = 0.f;
  for (int f = 0; f < F; ++f) { float v = p[f]; s += v; s2 += v * v; }
  float m = s / F;
  float var = fmaxf(s2 / F - m * m, 0.f);
  float inv = 1.0f / sqrtf(var + 1e-5f);
  for (int f = 0; f < F; ++f) {
    float v = (p[f] - m) * inv;
    if (relu) v = fmaxf(v, 0.f);
    p[f] = v;
  }
}

// roi_align (1x1 out, win 3, aligned=True) on layer [2,128,32,32]; skip [2,10000,128]
__global__ void roi_kernel(const float* __restrict__ pre, const float* __restrict__ layer,
                           float* __restrict__ skip) {
  long id = (long)blockIdx.x * TPB + threadIdx.x;
  if (id >= 2L * 10000 * 128) return;
  int c = (int)(id % 128); long t = id / 128;
  int n = (int)(t % 10000); int b = (int)(t / 10000);
  float px = pre[((size_t)b * 10000 + n) * 2 + 0] * 32.f;
  float py = pre[((size_t)b * 10000 + n) * 2 + 1] * 32.f;
  float x1 = px - 1.5f, y1 = py - 1.5f;
  const float* img = layer + ((size_t)b * 128 + c) * 1024;
  float acc = 0.f;
  for (int jy = 0; jy < 3; ++jy) {
    float yy = y1 + jy + 0.5f;
    for (int jx = 0; jx < 3; ++jx) {
      float xx = x1 + jx + 0.5f;
      float valid = (yy >= -1.f && yy <= 32.f && xx >= -1.f && xx <= 32.f) ? 1.f : 0.f;
      float yc = fminf(fmaxf(yy, 0.f), 31.f);
      float xc = fminf(fmaxf(xx, 0.f), 31.f);
      int y0 = (int)floorf(yc), x0 = (int)floorf(xc);
      int y1i = min(y0 + 1, 31), x1i = min(x0 + 1, 31);
      float ly = yc - (float)y0, lx = xc - (float)x0;
      float v = (1.f - ly) * (1.f - lx) * img[y0 * 32 + x0] + (1.f - ly) * lx * img[y0 * 32 + x1i]
              + ly * (1.f - lx) * img[y1i * 32 + x0] + ly * lx * img[y1i * 32 + x1i];
      acc += v * valid;
    }
  }
  skip[id] = acc * (1.f / 9.f);
}

// g146 = concat([g16, skip128, pre2], axis=2), [2,10000,146]
__global__ void concat146_kernel(const float* __restrict__ g16, const float* __restrict__ skip,
                                 const float* __restrict__ pre, float* __restrict__ out) {
  long id = (long)blockIdx.x * TPB + threadIdx.x;
  if (id >= 2L * 10000 * 146) return;
  int f = (int)(id % 146); long t = id / 146;
  int n = (int)(t % 10000); int b = (int)(t / 10000);
  size_t r = (size_t)b * 10000 + n;
  float v;
  if (f < 16) v = g16[r * 16 + f];
  else if (f < 144) v = skip[r * 128 + (f - 16)];
  else v = pre[r * 2 + (f - 144)];
  out[id] = v;
}

// gout[b,n0,f] = sum_k gin[b, up_idx[n0,k], f] * up_w[n0,k], F=32
__global__ void upsample_kernel(const float* __restrict__ gin, const int* __restrict__ up_idx,
                                const float* __restrict__ up_w, float* __restrict__ gout) {
  long id = (long)blockIdx.x * TPB + threadIdx.x;
  if (id >= 2L * 20000 * 32) return;
  int f = (int)(id % 32); long t = id / 32;
  int n0 = (int)(t % 20000); int b = (int)(t / 20000);
  float s = 0.f;
  for (int k = 0; k < 3; ++k) {
    int src = up_idx[n0 * 3 + k];
    s += gin[((size_t)b * 10000 + src) * 32 + f] * up_w[n0 * 3 + k];
  }
  gout[id] = s;
}

// ---------------- host orchestration ----------------
extern "C" void kernel_launch(void* const* d_in, const int* in_sizes, int n_in,
                              void* d_out, int out_size, void* d_ws, size_t ws_size,
                              hipStream_t stream) {
  (void)n_in; (void)out_size; (void)ws_size;
  // inputs (setup_inputs insertion order)
  const float* x = (const float*)d_in[0];
  const float* fc_w = (const float*)d_in[37];
  const float* fc_b = (const float*)d_in[38];
  const float* dec_w = (const float*)d_in[39];
  const float* dec_b = (const float*)d_in[40];
  const float* W6 = (const float*)d_in[41];
  const float* W5 = (const float*)d_in[43];
  const float* Wpre = (const float*)d_in[45];
  const float* W4 = (const float*)d_in[46];
  const float* W3 = (const float*)d_in[48];
  const float* W2 = (const float*)d_in[50];
  const float* Wlast = (const float*)d_in[52];
  const float* up_w = (const float*)d_in[53];
  const int* edges_c = (const int*)d_in[54];
  const int* edges_f = (const int*)d_in[55];
  const int* up_idx = (const int*)d_in[56];
  const int Ec = in_sizes[54] / 2, Ef = in_sizes[55] / 2;
  const int N1 = 10000, N0 = 20000;

  // workspace layout (floats)
  float* ws = (float*)d_ws;
  float* P1 = ws;                      // 16,777,216
  float* P2 = ws + 16777216L;          // 16,777,216
  float* P0 = ws + 33554432L;          // 4,194,304
  float* layerBuf = ws + 37748736L;    // 262,144 (conv6 output / flat)
  // aliases used after the encoder finishes:
  float* Tx0 = P1;                     // 3,000,000 each
  float* Tx1 = P1 + 3000000L;
  float* Tx2 = P1 + 6000000L;
  float* g146 = P1 + 9000000L;         // 2,920,000
  float* gA = P2;                      // 1,280,000 max
  float* gB = P2 + 1500000L;           // 1,280,000 max
  float* skipBuf = P2 + 3000000L;      // 2,560,000
  float* g0 = P2 + 5700000L;           // 320,000
  float* muBuf = P2 + 6100000L;        // 128
  float* dinvC = P2 + 6101000L;        // 10,000
  float* dinvF = P2 + 6112000L;        // 20,000
  float* degT = P2 + 6133000L;         // 20,000

  // ---- encoder: 6 residual blocks with 2x2 maxpool between ----
  const int chin[6] = {1, 8, 16, 32, 64, 128};
  const int chout[6] = {8, 16, 32, 64, 128, 128};
  int S = 1024;
  const float* cur = x;
  for (int i = 0; i < 6; ++i) {
    const float* w1 = (const float*)d_in[1 + 6 * i];
    const float* b1 = (const float*)d_in[2 + 6 * i];
    const float* w2 = (const float*)d_in[3 + 6 * i];
    const float* b2 = (const float*)d_in[4 + 6 * i];
    const float* wsk = (const float*)d_in[5 + 6 * i];
    const float* bsk = (const float*)d_in[6 + 6 * i];
    long n = 2L * chout[i] * S * S;
    conv3x3_kernel<<<gblocks(n), TPB, 0, stream>>>(cur, w1, b1, P1, chin[i], chout[i], S, 1);
    conv3x3_kernel<<<gblocks(n), TPB, 0, stream>>>(P1, w2, b2, P2, chout[i], chout[i], S, 0);
    float* dstbuf = (i == 5) ? layerBuf : P1;
    skip1x1_kernel<<<gblocks(n), TPB, 0, stream>>>(cur, P2, wsk, bsk, dstbuf, chin[i], chout[i], S);
    if (i < 5) {
      long np = 2L * chout[i] * (S / 2) * (S / 2);
      pool2_kernel<<<gblocks(np), TPB, 0, stream>>>(P1, P0, chout[i], S);
      cur = P0;
      S /= 2;
    }
  }

  // ---- fc_mu + decoder ----
  dim3 fcg(64, 2);
  fc_kernel<<<fcg, TPB, 0, stream>>>(layerBuf, fc_w, fc_b, muBuf, 131072);
  decoder_kernel<<<gblocks(160000), TPB, 0, stream>>>(dec_w, dec_b, muBuf, g0, 160000);

  // ---- degree-based symmetric norm factors ----
  fill_kernel<<<gblocks(N1), TPB, 0, stream>>>(degT, 0.f, N1);
  deg_kernel<<<gblocks(Ec), TPB, 0, stream>>>(edges_c + Ec, Ec, degT);
  dinv_kernel<<<gblocks(N1), TPB, 0, stream>>>(degT, dinvC, N1);
  fill_kernel<<<gblocks(N0), TPB, 0, stream>>>(degT, 0.f, N0);
  deg_kernel<<<gblocks(Ef), TPB, 0, stream>>>(edges_f + Ef, Ef, degT);
  dinv_kernel<<<gblocks(N0), TPB, 0, stream>>>(degT, dinvF, N0);

  // ---- ChebConv layer helper ----
  auto gemm = [&](const float* A, const float* Wp, float* C, int M, int F, int G) {
    dim3 g((M + 16 * MT - 1) / (16 * MT), (G + 15) / 16);
    wmma_gemm_acc<<<g, 32, 0, stream>>>(A, Wp, C, M, F, G);
  };
  auto cheb = [&](const float* gin, const float* Wc, int Kt, int F, int G, int N,
                  const int* src, const int* dst, int E, const float* dinv, float* outp) {
    long M = 2L * N;
    fill_kernel<<<gblocks(M * G), TPB, 0, stream>>>(outp, 0.f, M * G);
    copy_kernel<<<gblocks(M * F), TPB, 0, stream>>>(Tx0, gin, M * F);
    gemm(Tx0, Wc, outp, (int)M, F, G);
    if (Kt > 1) {
      fill_kernel<<<gblocks(M * F), TPB, 0, stream>>>(Tx1, 0.f, M * F);
      prop_kernel<<<gblocks((long)E * F), TPB, 0, stream>>>(Tx0, Tx1, src, dst, E, dinv, F, N, 1.f);
      gemm(Tx1, Wc + (size_t)F * G, outp, (int)M, F, G);
      float* Ta = Tx0; float* Tb = Tx1; float* Tc = Tx2;
      for (int k = 2; k < Kt; ++k) {
        negcopy_kernel<<<gblocks(M * F), TPB, 0, stream>>>(Tc, Ta, M * F);   // Tc = -T_{k-2}
        prop_kernel<<<gblocks((long)E * F), TPB, 0, stream>>>(Tb, Tc, src, dst, E, dinv, F, N, 2.f);
        gemm(Tc, Wc + (size_t)k * F * G, outp, (int)M, F, G);
        float* t = Ta; Ta = Tb; Tb = Tc; Tc = t;
      }
    }
  };

  const int* sc = edges_c; const int* dc = edges_c + Ec;
  const int* sf = edges_f; const int* df = edges_f + Ef;
  float* outFinal = (float*)d_out;             // [2,20000,2]
  float* preOut = (float*)d_out + 80000;       // [2,10000,2]

  // L6, L5 (coarse, F=G=16)
  cheb(g0, W6, 6, 16, 16, N1, sc, dc, Ec, dinvC, gA);
  inorm_kernel<<<gblocks(2L * N1), TPB, 0, stream>>>(gA, 2L * N1, 16, 1);
  cheb(gA, W5, 6, 16, 16, N1, sc, dc, Ec, dinvC, gB);
  inorm_kernel<<<gblocks(2L * N1), TPB, 0, stream>>>(gB, 2L * N1, 16, 1);
  // pre = cheb(g, Wpre) (K=1, G=2), written straight into d_out tail
  cheb(gB, Wpre, 1, 16, 2, N1, sc, dc, Ec, dinvC, preOut);
  // roi lookup on conv6 feature map, then concat to 146 features
  roi_kernel<<<gblocks(2L * N1 * 128), TPB, 0, stream>>>(preOut, layerBuf, skipBuf);
  concat146_kernel<<<gblocks(2L * N1 * 146), TPB, 0, stream>>>(gB, skipBuf, preOut, g146);
  // L4 (coarse, F=146 -> G=32)
  cheb(g146, W4, 6, 146, 32, N1, sc, dc, Ec, dinvC, gA);
  inorm_kernel<<<gblocks(2L * N1), TPB, 0, stream>>>(gA, 2L * N1, 32, 1);
  // sparse upsample N1 -> N0
  upsample_kernel<<<gblocks(2L * N0 * 32), TPB, 0, stream>>>(gA, up_idx, up_w, gB);
  // L3, L2 (fine, F=G=32)
  cheb(gB, W3, 6, 32, 32, N0, sf, df, Ef, dinvF, gA);
  inorm_kernel<<<gblocks(2L * N0), TPB, 0, stream>>>(gA, 2L * N0, 32, 1);
  cheb(gA, W2, 6, 32, 32, N0, sf, df, Ef, dinvF, gB);
  inorm_kernel<<<gblocks(2L * N0), TPB, 0, stream>>>(gB, 2L * N0, 32, 1);
  // final head (K=1, G=2) -> d_out head
  cheb(gB, Wlast, 1, 32, 2, N0, sf, df, Ef, dinvF, outFinal);
}